// ResRGCN_5437428597387
// MI455X (gfx1250) — compile-verified
//
#include <hip/hip_runtime.h>
#include <hip/hip_bf16.h>

#define HDIM 64
#define RREL 8
#define NBASE 4

typedef __attribute__((ext_vector_type(2))) float v2f;
typedef __attribute__((ext_vector_type(8))) float v8f;

// ---------------------------------------------------------------------------
// C[nrows x 64] = A[nrows x 64] @ W[64 x 64] (+ bias[64] if bias != null)
// Uses V_WMMA_F32_16X16X4_F32.  Block: 256 thr = 8 waves; each wave owns one
// 16x16 output tile (2 row-tiles x 4 col-tiles = 32x64 per block).
// W and a 32x64 A-strip are staged in LDS.
// ---------------------------------------------------------------------------
__global__ __launch_bounds__(256)
void mm64_wmma_kernel(const float* __restrict__ A, const float* __restrict__ W,
                      const float* __restrict__ bias, float* __restrict__ C,
                      int nrows)
{
    __shared__ float sW[64 * 64];   // 16 KB
    __shared__ float sA[32 * 64];   // 8 KB
    const int tid  = threadIdx.x;
    const int row0 = blockIdx.x * 32;

    // cooperative load of W (4096 floats) via float4
    for (int i = tid * 4; i < 64 * 64; i += 256 * 4)
        *(float4*)&sW[i] = *(const float4*)&W[i];
    // cooperative load of the 32x64 A strip
    for (int i = tid * 4; i < 32 * 64; i += 256 * 4) {
        int r = row0 + (i >> 6);
        float4 v = make_float4(0.f, 0.f, 0.f, 0.f);
        if (r < nrows) v = *(const float4*)&A[(size_t)r * 64 + (i & 63)];
        *(float4*)&sA[i] = v;
    }
    __syncthreads();

    const int wave = tid >> 5;        // 0..7
    const int lane = tid & 31;
    const int tm   = wave >> 2;       // row-tile 0..1
    const int tn   = wave & 3;        // col-tile 0..3
    const int hi   = lane >> 4;       // half-wave: K offset 0 / 2
    const int lm   = lane & 15;

    v8f c;
    const float bv = bias ? bias[tn * 16 + lm] : 0.0f;
#pragma unroll
    for (int i = 0; i < 8; ++i) c[i] = bv;

    const int arow = tm * 16 + lm;    // local row in sA (lanes 0-15 / 16-31 same M)
#pragma unroll
    for (int k0 = 0; k0 < 16; ++k0) {
        const int k = k0 * 4 + 2 * hi;
        // A frag: VGPR0 <-> K=k, VGPR1 <-> K=k+1 (per 32-bit A 16x4 layout)
        v2f a; a.x = sA[arow * 64 + k];           a.y = sA[arow * 64 + k + 1];
        // B frag: VGPR0 <-> row K=k col N=lm, VGPR1 <-> row K=k+1
        v2f b; b.x = sW[k * 64 + tn * 16 + lm];   b.y = sW[(k + 1) * 64 + tn * 16 + lm];
        c = __builtin_amdgcn_wmma_f32_16x16x4_f32(false, a, false, b,
                                                  (short)0, c, false, false);
    }

    // C/D layout: VGPR i, lanes 0-15 -> M=i, lanes 16-31 -> M=i+8; N = lm
#pragma unroll
    for (int i = 0; i < 8; ++i) {
        int r = row0 + tm * 16 + i + 8 * hi;
        if (r < nrows) C[(size_t)r * 64 + tn * 16 + lm] = c[i];
    }
}

// ---------------------------------------------------------------------------
// W_r = sum_b comp[r,b] * bases[b]   (one layer's 8 relation matrices)
// ---------------------------------------------------------------------------
__global__ __launch_bounds__(256)
void make_w_kernel(const float* __restrict__ bases, const float* __restrict__ comp,
                   float* __restrict__ Wcat)
{
    int idx = blockIdx.x * 256 + threadIdx.x;       // r*4096 + i*64 + o
    if (idx >= RREL * HDIM * HDIM) return;
    int r  = idx >> 12;
    int io = idx & 4095;
    float s = 0.f;
#pragma unroll
    for (int b = 0; b < NBASE; ++b)
        s += comp[r * NBASE + b] * bases[b * HDIM * HDIM + io];
    Wcat[idx] = s;
}

// ---------------------------------------------------------------------------
// per-(relation, dst) edge counts -> inverse means
// ---------------------------------------------------------------------------
__global__ __launch_bounds__(256)
void zero_u32_kernel(unsigned* __restrict__ p, int n)
{
    int i = blockIdx.x * 256 + threadIdx.x;
    if (i < n) p[i] = 0u;
}

__global__ __launch_bounds__(256)
void count_edges_kernel(const int* __restrict__ et, const int* __restrict__ dst,
                        unsigned* __restrict__ cnt, int E, int N)
{
    int e = blockIdx.x * 256 + threadIdx.x;
    if (e >= E) return;
    int r = et[e];
    atomicAdd(&cnt[(size_t)r * N + dst[e]], 1u);
}

__global__ __launch_bounds__(256)
void inv_kernel(const unsigned* __restrict__ cnt, float* __restrict__ inv, int n)
{
    int i = blockIdx.x * 256 + threadIdx.x;
    if (i >= n) return;
    unsigned c = cnt[i];
    inv[i] = 1.0f / (float)(c > 1u ? c : 1u);
}

// ---------------------------------------------------------------------------
// scatter for relation r: out[dst] += hr[src] * inv_r[dst]
// One wave screens 32 edges via ballot, then the full wave (float2/lane)
// processes each matching edge.
// ---------------------------------------------------------------------------
__global__ __launch_bounds__(256)
void scatter_rel_kernel(const int* __restrict__ et, const int* __restrict__ src,
                        const int* __restrict__ dst, const float* __restrict__ hr,
                        const float* __restrict__ inv_r, float* __restrict__ out,
                        int r, int E)
{
    const int lane = threadIdx.x & 31;
    const int wid  = (blockIdx.x * blockDim.x + threadIdx.x) >> 5;
    const int e0   = wid * 32;
    if (e0 >= E) return;
    const int e    = e0 + lane;
    const bool m   = (e < E) && (et[e] == r);
    unsigned long long mask = __ballot(m);   // wave32: bits [31:0]
    while (mask) {
        int b = __ffsll((long long)mask) - 1;
        mask &= mask - 1;
        int ee = e0 + b;
        int s  = src[ee];
        int d  = dst[ee];
        float w  = inv_r[d];
        float2 v = *(const float2*)&hr[(size_t)s * 64 + lane * 2];
        unsafeAtomicAdd(&out[(size_t)d * 64 + lane * 2 + 0], v.x * w);
        unsafeAtomicAdd(&out[(size_t)d * 64 + lane * 2 + 1], v.y * w);
    }
}

// ---------------------------------------------------------------------------
// h += relu(hn)
// ---------------------------------------------------------------------------
__global__ __launch_bounds__(256)
void relu_add_kernel(float* __restrict__ h, const float* __restrict__ hn, int n)
{
    int i = blockIdx.x * 256 + threadIdx.x;
    if (i < n) h[i] += fmaxf(hn[i], 0.0f);
}

// ---------------------------------------------------------------------------
// out[n,2] = h[n,:] @ lin2_w[64,2] + lin2_b
// ---------------------------------------------------------------------------
__global__ __launch_bounds__(256)
void lin2_kernel(const float* __restrict__ h, const float* __restrict__ w,
                 const float* __restrict__ b, float* __restrict__ out, int n)
{
    int i = blockIdx.x * 256 + threadIdx.x;
    if (i >= n) return;
    float a0 = b[0], a1 = b[1];
    const float* row = h + (size_t)i * 64;
#pragma unroll
    for (int k = 0; k < 64; ++k) {
        float v = row[k];
        a0 += v * w[k * 2 + 0];
        a1 += v * w[k * 2 + 1];
    }
    out[i * 2 + 0] = a0;
    out[i * 2 + 1] = a1;
}

// ---------------------------------------------------------------------------
extern "C" void kernel_launch(void* const* d_in, const int* in_sizes, int n_in,
                              void* d_out, int out_size, void* d_ws, size_t ws_size,
                              hipStream_t stream)
{
    const float* x        = (const float*)d_in[0];
    const int*   ei       = (const int*)  d_in[1];   // [2,E]
    const int*   et       = (const int*)  d_in[2];   // [E]
    const float* lin1_w   = (const float*)d_in[3];
    const float* lin1_b   = (const float*)d_in[4];
    const float* bases    = (const float*)d_in[5];   // [L,B,H,H]
    const float* comp     = (const float*)d_in[6];   // [L,R,B]
    const float* root     = (const float*)d_in[7];   // [L,H,H]
    const float* cbias    = (const float*)d_in[8];   // [L,H]
    const float* lin2_w   = (const float*)d_in[9];
    const float* lin2_b   = (const float*)d_in[10];
    float*       out      = (float*)d_out;

    const int N = in_sizes[0] / HDIM;
    const int E = in_sizes[2];
    const int* src = ei;
    const int* dst = ei + E;

    // workspace carve-up
    char* ws = (char*)d_ws;
    float* h0   = (float*)ws;                 ws += (size_t)N * HDIM * sizeof(float);
    float* h1   = (float*)ws;                 ws += (size_t)N * HDIM * sizeof(float);
    float* hr   = (float*)ws;                 ws += (size_t)N * HDIM * sizeof(float);
    float* inv  = (float*)ws;                 ws += (size_t)RREL * N * sizeof(float);
    unsigned* cnt = (unsigned*)ws;            ws += (size_t)RREL * N * sizeof(unsigned);
    float* Wcat = (float*)ws;                 ws += (size_t)RREL * HDIM * HDIM * sizeof(float);

    const int RN = RREL * N;
    dim3 blk(256);

    // per-(relation,dst) inverse mean weights (layer-invariant)
    zero_u32_kernel<<<(RN + 255) / 256, blk, 0, stream>>>(cnt, RN);
    count_edges_kernel<<<(E + 255) / 256, blk, 0, stream>>>(et, dst, cnt, E, N);
    inv_kernel<<<(RN + 255) / 256, blk, 0, stream>>>(cnt, inv, RN);

    const int mmGrid = (N + 31) / 32;

    // h0 = x @ lin1_w + lin1_b
    mm64_wmma_kernel<<<mmGrid, blk, 0, stream>>>(x, lin1_w, lin1_b, h0, N);

    // Layers 0 and 1 only: layer 2's output is discarded by the reference
    // (h is updated only when i < L-1), so it is dead code.
    for (int l = 0; l < 2; ++l) {
        make_w_kernel<<<(RREL * HDIM * HDIM + 255) / 256, blk, 0, stream>>>(
            bases + (size_t)l * NBASE * HDIM * HDIM, comp + (size_t)l * RREL * NBASE, Wcat);
        // h1 = h0 @ root_l + conv_bias_l
        mm64_wmma_kernel<<<mmGrid, blk, 0, stream>>>(
            h0, root + (size_t)l * HDIM * HDIM, cbias + (size_t)l * HDIM, h1, N);
        for (int r = 0; r < RREL; ++r) {
            // hr = h0 @ W_r
            mm64_wmma_kernel<<<mmGrid, blk, 0, stream>>>(
                h0, Wcat + (size_t)r * HDIM * HDIM, nullptr, hr, N);
            // h1[dst] += hr[src] * inv_r[dst]  over edges of type r
            scatter_rel_kernel<<<(E + 255) / 256, blk, 0, stream>>>(
                et, src, dst, hr, inv + (size_t)r * N, h1, r, E);
        }
        // h0 = h0 + relu(h1)
        relu_add_kernel<<<((N * HDIM) + 255) / 256, blk, 0, stream>>>(h0, h1, N * HDIM);
    }

    // out = h0 @ lin2_w + lin2_b
    lin2_kernel<<<(N + 255) / 256, blk, 0, stream>>>(h0, lin2_w, lin2_b, out, N);
}